// graph_network_action_44263932952755
// MI455X (gfx1250) — compile-verified
//
#include <hip/hip_runtime.h>
#include <hip/hip_bf16.h>

typedef __attribute__((ext_vector_type(2))) float v2f;
typedef __attribute__((ext_vector_type(4))) float v4f;
typedef __attribute__((ext_vector_type(8))) float v8f;

#define HIDDEN 256
#define ROWS_TOTAL (64 * 4096)          // NODE_NUM * N
#define TILE_ROWS 16
#define WAVES_PER_BLOCK 8

// out[i] = dot(h[i, 0:256], W[type[i/4096], :, 0]) + b[type]
// One wave per 16-row tile (contiguous 16KB of h). Bulk accumulation in f32
// VALU on NT b128 loads (memory-bound path), cross-lane reduction via 8
// chained V_WMMA_F32_16X16X4_F32 with an all-ones B and bias-broadcast C.
__global__ __launch_bounds__(256) void head_gemv_wmma_kernel(
    const float* __restrict__ h,
    const float* __restrict__ W,          // [8, 256, 1]
    const float* __restrict__ b,          // [8, 1]
    const int*   __restrict__ node_type,  // [64]
    float* __restrict__ out, int out_n)
{
  const int lane = threadIdx.x & 31;
  const int wave = threadIdx.x >> 5;
  const long tile = (long)blockIdx.x * WAVES_PER_BLOCK + wave;
  const long row_base = tile * TILE_ROWS;

  // 4096 rows per node; a 16-row tile never crosses a node boundary.
  const int node = (int)(row_base >> 12);
  const int type = node_type[node];
  const float* wt = W + (long)type * HIDDEN;
  const float bias = b[type];

  // Per-lane weight chunks: even half-row (cols 4L..4L+3) and odd half-row
  // (cols 128+4L..). Coalesced across the wave, loaded once per tile.
  const v4f wA = *(const v4f*)(wt + lane * 4);
  const v4f wB = *(const v4f*)(wt + 128 + lane * 4);

  const float* tp = h + row_base * HIDDEN;   // contiguous 16KB tile

  // Chunk c = i*32 + lane -> row = c>>6 (uniform per i), cols = (c&63)*4.
  float acc[TILE_ROWS];
#pragma unroll
  for (int r = 0; r < TILE_ROWS; ++r) {
    const v4f he = __builtin_nontemporal_load(
        (const v4f*)(tp + ((2 * r) * 32 + lane) * 4));       // cols 4L..
    const v4f ho = __builtin_nontemporal_load(
        (const v4f*)(tp + ((2 * r + 1) * 32 + lane) * 4));   // cols 128+4L..
    acc[r] = he.x * wA.x + he.y * wA.y + he.z * wA.z + he.w * wA.w
           + ho.x * wB.x + ho.y * wB.y + ho.z * wB.z + ho.w * wB.w;
  }

  // Transpose the 16x32 partial matrix through per-wave LDS (2KB/wave).
  __shared__ float lds[WAVES_PER_BLOCK * 32 * TILE_ROWS];
  float* wl = lds + wave * (32 * TILE_ROWS);
#pragma unroll
  for (int j = 0; j < 4; ++j) {
    v4f t;
    t.x = acc[4 * j + 0]; t.y = acc[4 * j + 1];
    t.z = acc[4 * j + 2]; t.w = acc[4 * j + 3];
    *(v4f*)(wl + lane * TILE_ROWS + 4 * j) = t;   // P[r][L] at wl[L*16 + r]
  }
  __syncthreads();

  // A-fragment (16x4 f32) layout: low lanes hold K={4k,4k+1}, high lanes
  // K={4k+2,4k+3}, M = lane&15. B = all-ones => every column of D is the
  // row sum; C initialized to bias broadcasts the +b.
  const int m  = lane & 15;
  const int j0 = (lane >> 4) << 1;   // 0 (low half) or 2 (high half)

  v8f d;
#pragma unroll
  for (int v = 0; v < 8; ++v) d[v] = bias;
  v2f ones; ones.x = 1.0f; ones.y = 1.0f;

#pragma unroll
  for (int k = 0; k < 8; ++k) {
    v2f a;
    a.x = wl[(4 * k + j0    ) * TILE_ROWS + m];
    a.y = wl[(4 * k + j0 + 1) * TILE_ROWS + m];
    d = __builtin_amdgcn_wmma_f32_16x16x4_f32(
        /*neg_a=*/false, a, /*neg_b=*/false, ones,
        /*c_mod=*/(short)0, d, /*reuse_a=*/false, /*reuse_b=*/false);
  }

  // Lane 0 holds rows 0..7 (N=0 column), lane 16 holds rows 8..15.
  if (lane == 0 || lane == 16) {
    const long base = row_base + ((lane == 16) ? 8 : 0);
    if (row_base + TILE_ROWS <= (long)out_n) {
      v4f lo; lo.x = d[0]; lo.y = d[1]; lo.z = d[2]; lo.w = d[3];
      v4f hi; hi.x = d[4]; hi.y = d[5]; hi.z = d[6]; hi.w = d[7];
      *(v4f*)(out + base)     = lo;
      *(v4f*)(out + base + 4) = hi;
    } else {
      // Last tile: reference drops the final element.
#pragma unroll
      for (int v = 0; v < 8; ++v)
        if (base + v < (long)out_n) out[base + v] = d[v];
    }
  }
}

extern "C" void kernel_launch(void* const* d_in, const int* in_sizes, int n_in,
                              void* d_out, int out_size, void* d_ws, size_t ws_size,
                              hipStream_t stream) {
  (void)in_sizes; (void)n_in; (void)d_ws; (void)ws_size;
  const float* h  = (const float*)d_in[0];
  const float* W  = (const float*)d_in[1];
  const float* b  = (const float*)d_in[2];
  const int* node_type = (const int*)d_in[3];
  float* out = (float*)d_out;

  const int tiles  = ROWS_TOTAL / TILE_ROWS;          // 16384
  const int blocks = tiles / WAVES_PER_BLOCK;         // 2048
  head_gemv_wmma_kernel<<<blocks, 256, 0, stream>>>(h, W, b, node_type, out, out_size);
}